// SegmentsRepr_43894565765274
// MI455X (gfx1250) — compile-verified
//
#include <hip/hip_runtime.h>
#include <hip/hip_bf16.h>
#include <math.h>

// Problem constants from the reference: frames [B=8, T=2048, D=256] fp32.
constexpr int BB = 8;
constexpr int T  = 2048;
constexpr int DD = 256;
constexpr int TC = 2047;   // length of cos / d / p2 arrays

typedef __attribute__((ext_vector_type(2))) float v2f;
typedef __attribute__((ext_vector_type(8))) float v8f;

// Hardware tanh (V_TANH_F32 trans op) when the builtin exists; ocml otherwise.
#if __has_builtin(__builtin_amdgcn_tanhf)
__device__ __forceinline__ float fast_tanh(float x) { return __builtin_amdgcn_tanhf(x); }
#else
__device__ __forceinline__ float fast_tanh(float x) { return tanhf(x); }
#endif

// v_rcp_f32 (1 ulp) + one Newton-Raphson step instead of the ~11-op IEEE
// division sequence; operands here are well-conditioned positive floats.
#if __has_builtin(__builtin_amdgcn_rcpf)
__device__ __forceinline__ float fast_rcp(float x) {
    float r = __builtin_amdgcn_rcpf(x);
    return r * fmaf(-x, r, 2.0f);
}
#else
__device__ __forceinline__ float fast_rcp(float x) { return 1.0f / x; }
#endif

// ---------------------------------------------------------------------------
// Kernel 1: cosine similarity of adjacent frames.  One wave32 per t.
// grid = BB * 256 blocks of 256 threads (8 waves/block, wave handles one t)
// ---------------------------------------------------------------------------
__global__ __launch_bounds__(256) void k1_cos(const float* __restrict__ frames,
                                              float* __restrict__ cosv) {
    int lane = threadIdx.x & 31;
    int wave = threadIdx.x >> 5;
    int blk  = blockIdx.x;
    int b    = blk >> 8;                     // 256 blocks per batch
    int t    = ((blk & 255) << 3) + wave;    // 0..2047
    if (t >= TC) return;
    const float* f0 = frames + ((size_t)b * T + t) * DD;
    const float* f1 = f0 + DD;
    float dot = 0.f, n0 = 0.f, n1 = 0.f;
    #pragma unroll
    for (int d = lane; d < DD; d += 32) {
        float a = f0[d], c = f1[d];
        dot = fmaf(a, c, dot);
        n0  = fmaf(a, a, n0);
        n1  = fmaf(c, c, n1);
    }
    #pragma unroll
    for (int off = 16; off > 0; off >>= 1) {
        dot += __shfl_xor(dot, off);
        n0  += __shfl_xor(n0,  off);
        n1  += __shfl_xor(n1,  off);
    }
    if (lane == 0)
        cosv[b * TC + t] = dot / fmaxf(sqrtf(n0) * sqrtf(n1), 1e-6f);
}

// ---------------------------------------------------------------------------
// Kernel 2: per batch: min/max normalize -> d -> p2 -> b -> cumsum bc, num_fr
// grid = BB blocks of 256 threads (each owns 8 contiguous t for the scan)
// ---------------------------------------------------------------------------
__global__ __launch_bounds__(256) void k2_boundary(const float* __restrict__ cosv,
                                                   float* __restrict__ bc,
                                                   float* __restrict__ num_fr) {
    __shared__ float dl[T];      // cos, then d, in place
    __shared__ float bv[T];      // boundary vector
    __shared__ float red[256];
    __shared__ float wsum[256];
    int b   = blockIdx.x;
    int tid = threadIdx.x;

    float mn = 1e30f, mx = -1e30f;
    for (int i = tid; i < TC; i += 256) {
        float c = cosv[b * TC + i];
        dl[i] = c;
        mn = fminf(mn, c);
        mx = fmaxf(mx, c);
    }
    red[tid] = mn; __syncthreads();
    for (int off = 128; off > 0; off >>= 1) {
        if (tid < off) red[tid] = fminf(red[tid], red[tid + off]);
        __syncthreads();
    }
    mn = red[0]; __syncthreads();
    red[tid] = mx; __syncthreads();
    for (int off = 128; off > 0; off >>= 1) {
        if (tid < off) red[tid] = fmaxf(red[tid], red[tid + off]);
        __syncthreads();
    }
    mx = red[0]; __syncthreads();

    float inv = 1.0f / (mx - mn);
    for (int i = tid; i < TC; i += 256)
        dl[i] = 1.0f - (dl[i] - mn) * inv;
    __syncthreads();

    // b[0] = 1 ; b[t] = tanh(1e7 * p2[t-1]) (straight-through fwd value)
    // p2 zeroed at indices {0,1,TC-2,TC-1} so no circular wrap is ever read.
    for (int t = tid; t < T; t += 256) {
        float v;
        if (t == 0) v = 1.0f;
        else {
            int i = t - 1;
            if (i < 2 || i >= TC - 2) v = 0.0f;
            else {
                float di = dl[i];
                float p2 = fminf(fmaxf(di - dl[i + 2], 0.f),
                                 fmaxf(di - dl[i - 2], 0.f));
                v = fast_tanh(1e7f * p2);
            }
        }
        bv[t] = v;
    }
    __syncthreads();

    // inclusive scan: 8 sequential per thread + Hillis-Steele over 256 sums
    float loc[8];
    float s = 0.f;
    #pragma unroll
    for (int j = 0; j < 8; j++) { s += bv[tid * 8 + j]; loc[j] = s; }
    wsum[tid] = s; __syncthreads();
    for (int off = 1; off < 256; off <<= 1) {
        float v   = wsum[tid];
        float add = (tid >= off) ? wsum[tid - off] : 0.f;
        __syncthreads();
        wsum[tid] = v + add;
        __syncthreads();
    }
    float base = (tid > 0) ? wsum[tid - 1] : 0.f;
    #pragma unroll
    for (int j = 0; j < 8; j++)
        bc[b * T + tid * 8 + j] = base + loc[j];
    if (tid == 255) num_fr[b] = wsum[255];
}

// ---------------------------------------------------------------------------
// Kernel 3: colsum[b,s] = sum_t (1 - tanh(10*|cc[s] - bc[t]|))
// grid = BB*8 blocks of 256 threads; bc staged in LDS
// ---------------------------------------------------------------------------
__global__ __launch_bounds__(256) void k3_colsum(const float* __restrict__ bc,
                                                 const float* __restrict__ num_fr,
                                                 float* __restrict__ colsum) {
    __shared__ float bcl[T];
    int b   = blockIdx.x >> 3;
    int s0  = (blockIdx.x & 7) << 8;
    int tid = threadIdx.x;
    for (int i = tid; i < T; i += 256) bcl[i] = bc[b * T + i];
    __syncthreads();
    int s = s0 + tid;
    float nf = num_fr[b];
    float c  = (float)(s + 1);
    float cc = (c > nf) ? 0.f : c;
    float sum = 0.f;
    #pragma unroll 8
    for (int t = 0; t < T; t++)
        sum += 1.0f - fast_tanh(10.0f * fabsf(cc - bcl[t]));
    colsum[b * T + s] = sum;
}

// ---------------------------------------------------------------------------
// Kernel 4: fused mask construction + fp32 WMMA GEMM.
// Block = (batch b, 16-row s-tile); 256 threads = 8 waves; wave w owns
// d-columns [32w, 32w+32) as two 16x16 f32 accumulators.
// Per 16-t chunk: async-stage frames tile directly into LDS
// (global_load_async_to_lds_b128, ASYNCcnt), compute 16x16 mask tile
// (one hw tanh + one hw rcp per thread), then 4x K=4
// v_wmma_f32_16x16x4_f32 per accumulator.
// ---------------------------------------------------------------------------
constexpr int FR_PITCH = 264;  // 2-row offset = 528 -> 16 banks: half-wave groups disjoint
constexpr int MK_PITCH = 18;   // even (float2 A loads), conflict-free column reads

__global__ __launch_bounds__(256) void k4_wmma_gemm(const float* __restrict__ frames,
                                                    const float* __restrict__ bc,
                                                    const float* __restrict__ num_fr,
                                                    const float* __restrict__ colsum,
                                                    float* __restrict__ out) {
    __shared__ __align__(16) float fr[16][FR_PITCH];  // frames tile [t_local][d]
    __shared__ __align__(16) float mk[16][MK_PITCH];  // mask tile   [s_local][t_local]

    int b   = blockIdx.x >> 7;          // 128 s-tiles per batch
    int s0  = (blockIdx.x & 127) << 4;
    int tid = threadIdx.x;
    int lane = tid & 31, wave = tid >> 5;
    int sl = tid >> 4;                  // 0..15 (also frames-tile row for loads)
    int tl = tid & 15;

    // s is fixed per thread for the whole block -> hoist cc, colsum
    float nf = num_fr[b];
    float c  = (float)(s0 + sl + 1);
    float cc = (c > nf) ? 0.f : c;
    float cs = colsum[b * T + s0 + sl];

    const float* fb  = frames + (size_t)b * T * DD;
    const float* bcb = bc + b * T;

    // Per-thread LDS destination (byte offset = low 32 bits of generic addr)
    unsigned ldsDst = (unsigned)(size_t)(&fr[sl][tl * 16]);

    v8f acc0 = {};
    v8f acc1 = {};
    int m     = lane & 15;              // A row / B column / C column within tile
    int kh    = lane >> 4;              // half-wave: K pair select (ISA 16x4 layout)
    int dbase = wave * 32;

    for (int t0 = 0; t0 < T; t0 += 16) {
        // async-stage frames[b, t0+sl, tl*16 .. +16) straight into LDS.
        // INST_OFFSET is added to BOTH the LDS and global addresses, so the
        // four 16B chunks share one LDS-base VGPR and one address pair.
        {
            const float* src = fb + (size_t)(t0 + sl) * DD + tl * 16;
            asm volatile(
                "global_load_async_to_lds_b128 %0, %1, off\n\t"
                "global_load_async_to_lds_b128 %0, %1, off offset:16\n\t"
                "global_load_async_to_lds_b128 %0, %1, off offset:32\n\t"
                "global_load_async_to_lds_b128 %0, %1, off offset:48"
                :: "v"(ldsDst), "v"(src) : "memory");
        }
        // mask tile element: one hw tanh + one hw rcp per thread
        {
            float u = fast_tanh(10.0f * fabsf(cc - bcb[t0 + tl]));
            mk[sl][tl] = (1.0f - u) * fast_rcp(cs + u);
        }
        // frames tile must be LDS-visible before the barrier releases readers
        asm volatile("s_wait_asynccnt 0x0" ::: "memory");
        __syncthreads();

        #pragma unroll
        for (int j = 0; j < 4; j++) {
            int k0 = 4 * j + 2 * kh;    // lanes 0-15: K0,K1 ; lanes 16-31: K2,K3
            v2f a = *(const v2f*)&mk[m][k0];
            v2f b0; b0.x = fr[k0][dbase + m];      b0.y = fr[k0 + 1][dbase + m];
            v2f b1; b1.x = fr[k0][dbase + 16 + m]; b1.y = fr[k0 + 1][dbase + 16 + m];
            acc0 = __builtin_amdgcn_wmma_f32_16x16x4_f32(false, a, false, b0,
                                                         (short)0, acc0, false, false);
            acc1 = __builtin_amdgcn_wmma_f32_16x16x4_f32(false, a, false, b1,
                                                         (short)0, acc1, false, false);
        }
        __syncthreads();
    }

    // C/D layout: lane<16 -> N=lane, M=vgpr ; lane>=16 -> N=lane-16, M=vgpr+8
    float* ob = out + ((size_t)b * T + s0) * DD;
    #pragma unroll
    for (int r = 0; r < 8; r++) {
        int mrow = r + 8 * kh;
        ob[(size_t)mrow * DD + dbase + m]      = acc0[r];
        ob[(size_t)mrow * DD + dbase + 16 + m] = acc1[r];
    }
}

// ---------------------------------------------------------------------------
extern "C" void kernel_launch(void* const* d_in, const int* in_sizes, int n_in,
                              void* d_out, int out_size, void* d_ws, size_t ws_size,
                              hipStream_t stream) {
    const float* frames = (const float*)d_in[0];
    float* out = (float*)d_out;
    float* ws  = (float*)d_ws;

    // workspace layout (floats): cos[8*2047] | bc[8*2048] | num_fr[8] | colsum[8*2048]
    float* cosv   = ws;
    float* bc     = ws + BB * T;         // padded past 8*2047
    float* nf     = bc + BB * T;
    float* colsum = nf + BB;

    k1_cos      <<<BB * 256, 256, 0, stream>>>(frames, cosv);
    k2_boundary <<<BB,       256, 0, stream>>>(cosv, bc, nf);
    k3_colsum   <<<BB * 8,   256, 0, stream>>>(bc, nf, colsum);
    k4_wmma_gemm<<<BB * 128, 256, 0, stream>>>(frames, bc, nf, colsum, out);
}